// NonLinearQuantizer_71708773974328
// MI455X (gfx1250) — compile-verified
//
#include <hip/hip_runtime.h>

typedef __attribute__((ext_vector_type(2))) float v2f;
typedef __attribute__((ext_vector_type(4))) float v4f;
typedef __attribute__((ext_vector_type(8))) float v8f;

#define DD   256      // feature dim
#define NPOS 8192     // B*H*W
#define KCB  8192     // codebook entries
#define HW   1024     // H*W
#define HARD_ELEMS (8 * 256 * 32 * 32)

// ---------------------------------------------------------------------------
// Transpose latent [B, D, H, W] -> q [N=B*H*W, D]  (coalesced writes)
// ---------------------------------------------------------------------------
__global__ __launch_bounds__(256) void k_transpose(const float* __restrict__ latent,
                                                   float* __restrict__ q) {
    int tid = blockIdx.x * 256 + threadIdx.x;          // tid = n*256 + d
    if (tid >= NPOS * DD) return;
    int d = tid & 255;
    int n = tid >> 8;
    int b = n >> 10;          // n / 1024
    int rem = n & 1023;       // h*32 + w
    q[tid] = latent[(size_t)(b * 256 + d) * HW + rem];
}

// ---------------------------------------------------------------------------
// C[m,n] = s * dot(A[m,:], B[n,:])  (C = A·Bᵀ), K = 256, via v_wmma_f32_16x16x4_f32
// WG = 256 threads = 8 waves arranged 2(M) x 4(N); wave tile = 64x64 (4x4 frags)
// WG tile = 128(M) x 256(N);  grid = (N/256, M/128)
//
// f32 WMMA fragment layouts (wave32):
//   A 16x4:  lane r=L&15 is row M=r; hi=L>>4 selects K pair {k,k+1} (hi=0)
//            or {k+2,k+3} (hi=1)  -> contiguous float2 at A[row*256 + k + 2*hi]
//   B 4x16:  lane r is column N=r; same K-pair selection -> contiguous float2
//            at B[col*256 + k + 2*hi]   (B supplied as rows of the rhs matrix)
//   C 16x16: vgpr v holds row M = v + 8*hi, col N = r
// ---------------------------------------------------------------------------
__global__ __launch_bounds__(256) void k_gemm_abt(const float* __restrict__ A,
                                                  const float* __restrict__ B,
                                                  float* __restrict__ C, int ldc,
                                                  const float* __restrict__ scale_ptr,
                                                  float scale_const) {
    const int lane = threadIdx.x & 31;
    const int wave = threadIdx.x >> 5;
    const int wm = wave >> 2;            // 0..1
    const int wn = wave & 3;             // 0..3
    const int r  = lane & 15;
    const int hi = lane >> 4;            // 0/1

    const int m_base = blockIdx.y * 128 + wm * 64;
    const int n_base = blockIdx.x * 256 + wn * 64;

    const float* ap = A + (size_t)(m_base + r) * DD + hi * 2;
    const float* bp = B + (size_t)(n_base + r) * DD + hi * 2;

    v8f c[4][4] = {};

#pragma unroll 2
    for (int k = 0; k < DD; k += 4) {
        v2f a[4], b[4];
#pragma unroll
        for (int i = 0; i < 4; ++i)
            a[i] = *reinterpret_cast<const v2f*>(ap + (size_t)(i * 16) * DD + k);
#pragma unroll
        for (int j = 0; j < 4; ++j)
            b[j] = *reinterpret_cast<const v2f*>(bp + (size_t)(j * 16) * DD + k);
#pragma unroll
        for (int i = 0; i < 4; ++i)
#pragma unroll
            for (int j = 0; j < 4; ++j)
                c[i][j] = __builtin_amdgcn_wmma_f32_16x16x4_f32(
                    false, a[i], false, b[j], (short)0, c[i][j], false, false);
    }

    float s = scale_const;
    if (scale_ptr) s *= scale_ptr[0];

    // store: frag (i,j), vgpr v -> row (m_base + i*16 + v + 8*hi), col (n_base + j*16 + r)
#pragma unroll
    for (int i = 0; i < 4; ++i) {
#pragma unroll
        for (int j = 0; j < 4; ++j) {
            const int col = n_base + j * 16 + r;
#pragma unroll
            for (int v = 0; v < 8; ++v) {
                const int row = m_base + i * 16 + v + hi * 8;
                C[(size_t)row * ldc + col] = c[i][j][v] * s;
            }
        }
    }
}

// ---------------------------------------------------------------------------
// Per-row argmax: true_code = argmax(logit), idx = argmax(logit + gumbel)
// One 256-thread WG per row; float4 streaming; first-index tie-break.
// ---------------------------------------------------------------------------
__global__ __launch_bounds__(256) void k_argmax(const float* __restrict__ logit,
                                                const float* __restrict__ gumbel,
                                                float* __restrict__ true_code,
                                                int* __restrict__ idx_out) {
    const int m = blockIdx.x;
    const int tid = threadIdx.x;
    const v4f* lrow = reinterpret_cast<const v4f*>(logit  + (size_t)m * KCB);
    const v4f* grow = reinterpret_cast<const v4f*>(gumbel + (size_t)m * KCB);

    float bv = -__builtin_huge_valf(); int bi = 0;   // plain logit
    float pv = -__builtin_huge_valf(); int pi = 0;   // gumbel-perturbed
#pragma unroll
    for (int it = 0; it < KCB / 1024; ++it) {
        int c4 = it * 256 + tid;        // float4 index; strictly increasing per thread
        v4f l = lrow[c4];
        v4f g = grow[c4];
        int col = c4 * 4;
#pragma unroll
        for (int e = 0; e < 4; ++e) {
            float lv = l[e];
            if (lv > bv) { bv = lv; bi = col + e; }
            float p = lv + g[e];
            if (p > pv) { pv = p; pi = col + e; }
        }
    }

    __shared__ float s_bv[256]; __shared__ int s_bi[256];
    __shared__ float s_pv[256]; __shared__ int s_pi[256];
    s_bv[tid] = bv; s_bi[tid] = bi;
    s_pv[tid] = pv; s_pi[tid] = pi;
    __syncthreads();
    for (int off = 128; off > 0; off >>= 1) {
        if (tid < off) {
            float ov = s_bv[tid + off]; int oi = s_bi[tid + off];
            if (ov > s_bv[tid] || (ov == s_bv[tid] && oi < s_bi[tid])) { s_bv[tid] = ov; s_bi[tid] = oi; }
            float qv = s_pv[tid + off]; int qi = s_pi[tid + off];
            if (qv > s_pv[tid] || (qv == s_pv[tid] && qi < s_pi[tid])) { s_pv[tid] = qv; s_pi[tid] = qi; }
        }
        __syncthreads();
    }
    if (tid == 0) {
        true_code[m] = (float)s_bi[0];
        idx_out[m]   = s_pi[0];
    }
}

// ---------------------------------------------------------------------------
// hard[b, d, h, w] = target[idx[n], d],  n = b*H*W + h*W + w
// One WG per position n; thread = d (coalesced target read).
// ---------------------------------------------------------------------------
__global__ __launch_bounds__(256) void k_gather(const float* __restrict__ target,
                                                const int* __restrict__ idx_in,
                                                float* __restrict__ hard) {
    const int n = blockIdx.x;
    const int d = threadIdx.x;
    const int idx = idx_in[n];
    float val = target[(size_t)idx * DD + d];
    int b = n >> 10;
    int rem = n & 1023;                       // h*32 + w
    hard[(size_t)(b * 256 + d) * HW + rem] = val;
}

// ---------------------------------------------------------------------------
extern "C" void kernel_launch(void* const* d_in, const int* in_sizes, int n_in,
                              void* d_out, int out_size, void* d_ws, size_t ws_size,
                              hipStream_t stream) {
    const float* latent   = (const float*)d_in[0];
    // d_in[1] = temperature (unused for forward values)
    const float* gumbel   = (const float*)d_in[2];
    const float* codebook = (const float*)d_in[3];
    const float* wk       = (const float*)d_in[4];
    const float* wv       = (const float*)d_in[5];
    const float* t1       = (const float*)d_in[6];

    float* out       = (float*)d_out;
    float* hard      = out;                       // [B, D, H, W] = 2,097,152
    float* true_code = out + HARD_ELEMS;          // [B, H, W]    = 8,192
    float* logit     = true_code + NPOS;          // [N, K]       = 67,108,864

    float* ws     = (float*)d_ws;
    float* q_ws   = ws;                           //  8 MB
    float* kproj  = q_ws  + (size_t)NPOS * DD;    //  8 MB
    float* target = kproj + (size_t)KCB * DD;     //  8 MB
    int*   idx_ws = (int*)(target + (size_t)KCB * DD);

    // 1) latent -> q [N, D]
    k_transpose<<<(NPOS * DD + 255) / 256, 256, 0, stream>>>(latent, q_ws);

    // 2) kproj = codebook @ wkᵀ ; target = codebook @ wvᵀ   (N=256 -> x=1, M=8192/128 -> y=64)
    k_gemm_abt<<<dim3(1, 64), 256, 0, stream>>>(codebook, wk, kproj, DD, nullptr, 1.0f);
    k_gemm_abt<<<dim3(1, 64), 256, 0, stream>>>(codebook, wv, target, DD, nullptr, 1.0f);

    // 3) logit = q @ kprojᵀ * t1 / sqrt(256)   (N=8192/256 -> x=32, M=8192/128 -> y=64)
    k_gemm_abt<<<dim3(32, 64), 256, 0, stream>>>(q_ws, kproj, logit, KCB, t1, 1.0f / 16.0f);

    // 4) argmaxes
    k_argmax<<<NPOS, 256, 0, stream>>>(logit, gumbel, true_code, idx_ws);

    // 5) hard = gather(target, idx)
    k_gather<<<NPOS, 256, 0, stream>>>(target, idx_ws, hard);
}